// Method_Classification_37821482008663
// MI455X (gfx1250) — compile-verified
//
#include <hip/hip_runtime.h>
#include <math.h>

// ---------------------------------------------------------------------------
// 2-layer GCN for MI455X (gfx1250, wave32).
//   L1: h1 = x @ W1            -> f32 WMMA 16x16x4 (full precision, HBM-bound)
//       agg1 = scatter_add(norm * h1[row] -> col)   (L2-resident atomics)
//       o1 = relu(agg1 + b1)
//   L2: h2 = o1 @ W2 (tiny)    -> VALU
//       agg2 = scatter_add(norm * h2[row] -> col)
//       out = sigmoid(agg2 + b2)
// ---------------------------------------------------------------------------

#define IN_DIM 768
#define HID    50
#define HPAD   64            // hidden padded to 4 WMMA N-tiles
#define OUTD   2
#define KPAIRS (IN_DIM / 2)  // 384 K-pairs
// LDS row stride per K-pair, in floats. 128 needed; pad to 160 so the
// khalf=1 half-wave (row +1) lands on banks 32..63 (stride 160 = 32 mod 64).
#define KPSTRIDE 160

typedef __attribute__((ext_vector_type(2))) float v2f;
typedef __attribute__((ext_vector_type(8))) float v8f;

// ---------------- degree / normalization ----------------------------------
__global__ void k_deg_init(float* __restrict__ deg, int n) {
  int i = blockIdx.x * blockDim.x + threadIdx.x;
  if (i < n) deg[i] = 1.0f;                 // self-loop contribution
}

__global__ void k_deg_count(const long long* __restrict__ col,
                            float* __restrict__ deg, int n_edges) {
  int i = blockIdx.x * blockDim.x + threadIdx.x;
  if (i < n_edges) atomicAdd(&deg[(int)col[i]], 1.0f);
}

__global__ void k_deg_rsqrt(const float* __restrict__ deg,
                            float* __restrict__ dis, int n) {
  int i = blockIdx.x * blockDim.x + threadIdx.x;
  if (i < n) dis[i] = rsqrtf(deg[i]);       // deg >= 1 always (self-loops)
}

// ---------------- GEMM1: [N,768] x [768,50] via V_WMMA_F32_16X16X4_F32 -----
// Block = 256 threads = 8 waves. Each wave owns a 16-node M-tile and computes
// a 16x64 output tile (4 v8f accumulators). W1 is weight-stationary in LDS in
// a K-pair-interleaved layout: w1s[(k>>1)*KPSTRIDE + n*2 + (k&1)], so each
// lane's B fragment (W1[ka][n], W1[ka+1][n]) is ONE contiguous ds_load_b64
// into an even-aligned VGPR pair (no repack movs before the WMMA).
__global__ __launch_bounds__(256) void k_gemm1_wmma(
    const float* __restrict__ x, const float* __restrict__ W1,
    float* __restrict__ h1, int n_nodes) {
  extern __shared__ float w1s[];            // KPAIRS * KPSTRIDE floats (245.8 KB)

  // cooperative staging: one float2 (both halves of a K-pair) per iteration
  for (int idx = threadIdx.x; idx < KPAIRS * HPAD; idx += 256) {
    int kp = idx >> 6;                      // K-pair index
    int n  = idx & (HPAD - 1);
    v2f w;
    w.x = (n < HID) ? W1[(2 * kp    ) * HID + n] : 0.0f;
    w.y = (n < HID) ? W1[(2 * kp + 1) * HID + n] : 0.0f;
    *(v2f*)(w1s + kp * KPSTRIDE + n * 2) = w;
  }
  __syncthreads();

  const int wave  = threadIdx.x >> 5;
  const int lane  = threadIdx.x & 31;
  const int mloc  = lane & 15;              // M (A) / N (B) position in tile
  const int khalf = lane >> 4;              // 0 -> K{0,1}, 1 -> K{2,3}

  const long long tile = (long long)blockIdx.x * 8 + wave;
  const long long m0   = tile * 16;
  long long mrow = m0 + mloc;
  if (mrow > (long long)n_nodes - 1) mrow = (long long)n_nodes - 1; // clamp OOB reads
  const float* xrow = x + mrow * IN_DIM;

  // per-lane LDS base: advance by 2 K-pair rows (one k-step of 4) per iter
  const float* wb = w1s + khalf * KPSTRIDE + mloc * 2;

  v8f acc0 = {}, acc1 = {}, acc2 = {}, acc3 = {};

#pragma unroll 4
  for (int k0 = 0; k0 < IN_DIM; k0 += 4) {
    const int ka = k0 + 2 * khalf;
    // A fragment: lane holds x[m][ka], x[m][ka+1]  (ISA A 16x4 f32 layout)
    v2f a = *(const v2f*)(xrow + ka);
    // B fragments: single b64 each, constant offsets for the 4 N-tiles
    const float* wp = wb + (k0 >> 1) * KPSTRIDE;
    v2f b0 = *(const v2f*)(wp);
    v2f b1 = *(const v2f*)(wp + 32);
    v2f b2 = *(const v2f*)(wp + 64);
    v2f b3 = *(const v2f*)(wp + 96);
    acc0 = __builtin_amdgcn_wmma_f32_16x16x4_f32(false, a, false, b0, (short)0, acc0, false, false);
    acc1 = __builtin_amdgcn_wmma_f32_16x16x4_f32(false, a, false, b1, (short)0, acc1, false, false);
    acc2 = __builtin_amdgcn_wmma_f32_16x16x4_f32(false, a, false, b2, (short)0, acc2, false, false);
    acc3 = __builtin_amdgcn_wmma_f32_16x16x4_f32(false, a, false, b3, (short)0, acc3, false, false);
  }

  // D layout: VGPR r -> M = r (lanes 0-15) / M = r+8 (lanes 16-31), N = mloc
#pragma unroll
  for (int r = 0; r < 8; ++r) {
    long long m = m0 + r + 8 * khalf;
    if (m < n_nodes) {
      float* hrow = h1 + m * HID;
      hrow[mloc] = acc0[r];
      if (mloc + 16 < HID) hrow[mloc + 16] = acc1[r];
      if (mloc + 32 < HID) hrow[mloc + 32] = acc2[r];
      if (mloc + 48 < HID) hrow[mloc + 48] = acc3[r];   // n = 48,49 valid
    }
  }
}

// ---------------- scatter layer 1: 16 lanes per message --------------------
__global__ void k_scatter1(const long long* __restrict__ row,
                           const long long* __restrict__ col,
                           const float* __restrict__ dis,
                           const float* __restrict__ h1,
                           float* __restrict__ agg1,
                           int n_edges, int n_nodes) {
  long long t = (long long)blockIdx.x * blockDim.x + threadIdx.x;
  long long e = t >> 4;
  int j0 = (int)(t & 15);
  long long total = (long long)n_edges + n_nodes;
  if (e >= total) return;
  long long r, c;
  if (e < n_edges) { r = row[e]; c = col[e]; }
  else             { r = c = e - n_edges; }          // self-loop
  float norm = dis[r] * dis[c];
  const float* hr = h1 + r * HID;
  float* ac = agg1 + c * HID;
#pragma unroll
  for (int j = j0; j < HID; j += 16)
    atomicAdd(&ac[j], hr[j] * norm);
}

// ---------------- bias + ReLU (in place on agg1) ---------------------------
__global__ void k_bias_relu(float* __restrict__ agg1,
                            const float* __restrict__ b1, long long total) {
  long long i = (long long)blockIdx.x * blockDim.x + threadIdx.x;
  if (i < total) {
    float v = agg1[i] + b1[(int)(i % HID)];
    agg1[i] = v > 0.0f ? v : 0.0f;
  }
}

// ---------------- GEMM2 (tiny): [N,50] x [50,2] ----------------------------
__global__ void k_gemm2(const float* __restrict__ o1,
                        const float* __restrict__ W2,
                        float* __restrict__ h2, int n_nodes) {
  int i = blockIdx.x * blockDim.x + threadIdx.x;
  if (i >= n_nodes) return;
  const float* r = o1 + (long long)i * HID;
  float a0 = 0.0f, a1 = 0.0f;
#pragma unroll
  for (int k = 0; k < HID; ++k) {
    float v = r[k];
    a0 += v * W2[k * OUTD + 0];
    a1 += v * W2[k * OUTD + 1];
  }
  h2[(long long)i * OUTD + 0] = a0;
  h2[(long long)i * OUTD + 1] = a1;
}

// ---------------- scatter layer 2: 1 thread per message --------------------
__global__ void k_scatter2(const long long* __restrict__ row,
                           const long long* __restrict__ col,
                           const float* __restrict__ dis,
                           const float* __restrict__ h2,
                           float* __restrict__ agg2,
                           int n_edges, int n_nodes) {
  long long e = (long long)blockIdx.x * blockDim.x + threadIdx.x;
  long long total = (long long)n_edges + n_nodes;
  if (e >= total) return;
  long long r, c;
  if (e < n_edges) { r = row[e]; c = col[e]; }
  else             { r = c = e - n_edges; }
  float norm = dis[r] * dis[c];
  atomicAdd(&agg2[c * OUTD + 0], h2[r * OUTD + 0] * norm);
  atomicAdd(&agg2[c * OUTD + 1], h2[r * OUTD + 1] * norm);
}

// ---------------- bias + sigmoid -> d_out ----------------------------------
__global__ void k_final(const float* __restrict__ agg2,
                        const float* __restrict__ b2,
                        float* __restrict__ out, long long total) {
  long long i = (long long)blockIdx.x * blockDim.x + threadIdx.x;
  if (i < total) {
    float v = agg2[i] + b2[(int)(i & 1)];
    out[i] = 1.0f / (1.0f + expf(-v));
  }
}

// ---------------------------------------------------------------------------
extern "C" void kernel_launch(void* const* d_in, const int* in_sizes, int n_in,
                              void* d_out, int out_size, void* d_ws, size_t ws_size,
                              hipStream_t stream) {
  const float*     x  = (const float*)d_in[0];
  const long long* ei = (const long long*)d_in[1];   // int64 [2, E] flat
  const float*     W1 = (const float*)d_in[2];
  const float*     b1 = (const float*)d_in[3];
  const float*     W2 = (const float*)d_in[4];
  const float*     b2 = (const float*)d_in[5];

  const int n_nodes = in_sizes[0] / IN_DIM;
  const int n_edges = in_sizes[1] / 2;
  const long long* row = ei;
  const long long* col = ei + n_edges;

  // workspace carve-out (~43 MB): deg | dis | h1 | agg1 | h2 | agg2
  const long long nb = n_nodes;
  float* deg  = (float*)d_ws;
  float* dis  = deg  + nb;
  float* h1   = dis  + nb;
  float* agg1 = h1   + nb * HID;
  float* h2   = agg1 + nb * HID;
  float* agg2 = h2   + nb * OUTD;

  hipMemsetAsync(agg1, 0, (size_t)nb * HID  * sizeof(float), stream);
  hipMemsetAsync(agg2, 0, (size_t)nb * OUTD * sizeof(float), stream);

  // D^{-1/2} with self-loops
  k_deg_init <<<(n_nodes + 255) / 256, 256, 0, stream>>>(deg, n_nodes);
  k_deg_count<<<(n_edges + 255) / 256, 256, 0, stream>>>(col, deg, n_edges);
  k_deg_rsqrt<<<(n_nodes + 255) / 256, 256, 0, stream>>>(deg, dis, n_nodes);

  // Layer 1 GEMM (WMMA f32): 8 M-tiles (16 rows each) per 256-thread block
  const int mtiles   = (n_nodes + 15) / 16;
  const int g1blocks = (mtiles + 7) / 8;
  const size_t lds_bytes = (size_t)KPAIRS * KPSTRIDE * sizeof(float); // 245.8 KB < 320 KB/WGP
  k_gemm1_wmma<<<g1blocks, 256, lds_bytes, stream>>>(x, W1, h1, n_nodes);

  const long long msgs = (long long)n_edges + n_nodes;
  const long long t1   = msgs * 16;
  k_scatter1<<<(unsigned)((t1 + 255) / 256), 256, 0, stream>>>(row, col, dis, h1, agg1,
                                                               n_edges, n_nodes);
  k_bias_relu<<<(unsigned)((nb * HID + 255) / 256), 256, 0, stream>>>(agg1, b1, nb * HID);

  // Layer 2 (tiny GEMM + scatter + sigmoid)
  k_gemm2<<<(n_nodes + 255) / 256, 256, 0, stream>>>(agg1, W2, h2, n_nodes);
  k_scatter2<<<(unsigned)((msgs + 255) / 256), 256, 0, stream>>>(row, col, dis, h2, agg2,
                                                                 n_edges, n_nodes);
  k_final<<<(unsigned)((nb * OUTD + 255) / 256), 256, 0, stream>>>(agg2, b2, (float*)d_out,
                                                                   nb * OUTD);
}